// CurriculumMining_40767829574574
// MI455X (gfx1250) — compile-verified
//
#include <hip/hip_runtime.h>

#define BN 8192
#define DK 1024

typedef __attribute__((ext_vector_type(2))) float v2f;
typedef __attribute__((ext_vector_type(8))) float v8f;

// b128 async-copy payload type, matching the builtin's expected signature:
// (int __attribute__((vector_size(16))) addrspace(1)*,  ... addrspace(3)*, imm, imm)
typedef int v4i __attribute__((vector_size(16)));
typedef __attribute__((address_space(1))) v4i* gv4i_p;
typedef __attribute__((address_space(3))) v4i* lv4i_p;

__device__ __forceinline__ v8f wmma4(v2f a, v2f b, v8f c) {
  // V_WMMA_F32_16X16X4_F32: D(16x16,f32) = A(16x4,f32) * B(4x16,f32) + C
  return __builtin_amdgcn_wmma_f32_16x16x4_f32(
      /*neg_a=*/false, a, /*neg_b=*/false, b,
      /*c_mod=*/(short)0, c, /*reuse_a=*/false, /*reuse_b=*/false);
}

// Order-preserving float -> uint key (total order incl. negatives).
__device__ __forceinline__ unsigned fkey(float f) {
  const unsigned u = __float_as_uint(f);
  return (u & 0x80000000u) ? ~u : (u | 0x80000000u);
}

// ---------------------------------------------------------------------------
// Row sum-of-squares: one wave (32 lanes) per row, 8 rows per 256-thread block
// ---------------------------------------------------------------------------
__global__ __launch_bounds__(256) void cm_rownorm_kernel(
    const float* __restrict__ X, float* __restrict__ out) {
  const int wave = threadIdx.x >> 5;
  const int lane = threadIdx.x & 31;
  const int row  = blockIdx.x * 8 + wave;
  const float* p = X + (size_t)row * DK;
  float s = 0.f;
  for (int c = lane; c < DK; c += 32) {
    float v = p[c];
    s = fmaf(v, v, s);
  }
#pragma unroll
  for (int off = 16; off > 0; off >>= 1) s += __shfl_xor(s, off, 32);
  if (lane == 0) out[row] = s;
}

// ---------------------------------------------------------------------------
// Distance GEMM via f32 WMMA.
// Wave computes a 32x32 tile of dists = sqrt(max(x2 + y2 - 2*T.S^T, 0)).
// Block = 8 waves arranged 2x4 -> 64(M) x 128(N) block tile.
// Fragment layout (wave32, 16x4 f32 A / 4x16 f32 B):
//   lane L: r = L&15, kh = (L>>4)*2 ; A frag = T[rowM + r][k+kh .. k+kh+1]
//                                     B frag = S[rowN + r][k+kh .. k+kh+1]
// C/D: vgpr v, lane L -> M = v + 8*(L>>4), N = L&15.
// ---------------------------------------------------------------------------
__global__ __launch_bounds__(256) void cm_dist_gemm_kernel(
    const float* __restrict__ T, const float* __restrict__ S,
    const float* __restrict__ x2, const float* __restrict__ y2,
    float* __restrict__ Dmat) {
  const int wave = threadIdx.x >> 5;
  const int lane = threadIdx.x & 31;
  const int wm = wave >> 2;      // 0..1
  const int wn = wave & 3;       // 0..3
  const int tileM0 = blockIdx.y * 64  + wm * 32;
  const int tileN0 = blockIdx.x * 128 + wn * 32;

  const int r  = lane & 15;
  const int kh = (lane >> 4) * 2;  // 0 or 2

  const float* A0 = T + (size_t)(tileM0 + r)      * DK + kh;
  const float* A1 = T + (size_t)(tileM0 + 16 + r) * DK + kh;
  const float* B0 = S + (size_t)(tileN0 + r)      * DK + kh;
  const float* B1 = S + (size_t)(tileN0 + 16 + r) * DK + kh;

  v8f acc00 = {}, acc01 = {}, acc10 = {}, acc11 = {};

#pragma unroll 4
  for (int k = 0; k < DK; k += 4) {
    v2f a0 = *(const v2f*)(A0 + k);
    v2f a1 = *(const v2f*)(A1 + k);
    v2f b0 = *(const v2f*)(B0 + k);
    v2f b1 = *(const v2f*)(B1 + k);
    acc00 = wmma4(a0, b0, acc00);
    acc01 = wmma4(a0, b1, acc01);
    acc10 = wmma4(a1, b0, acc10);
    acc11 = wmma4(a1, b1, acc11);
  }

  const int mhi = (lane >> 4) * 8;
  const int n   = lane & 15;
#pragma unroll
  for (int i = 0; i < 2; ++i) {
#pragma unroll
    for (int j = 0; j < 2; ++j) {
      const v8f acc = (i == 0) ? (j == 0 ? acc00 : acc01)
                               : (j == 0 ? acc10 : acc11);
      const int gn = tileN0 + j * 16 + n;
      const float ys = y2[gn];
#pragma unroll
      for (int v = 0; v < 8; ++v) {
        const int gm = tileM0 + i * 16 + mhi + v;
        const float d2 = x2[gm] + ys - 2.0f * acc[v];
        Dmat[(size_t)gm * BN + gn] = sqrtf(fmaxf(d2, 0.0f));
      }
    }
  }
}

// ---------------------------------------------------------------------------
// Per-row selection: one 256-thread block per row. The entire 32 KB row is
// copied into LDS with GLOBAL_LOAD_ASYNC_TO_LDS_B128 (ASYNCcnt path, bypasses
// VGPRs), with the diagonal overwritten to -1.0 (== rev_dists). Computes:
//  - hard_semi_negative: lexicographic argmin of (|d - diag|, j), j != i
//  - negative_sample_idx: column of the rank-idx_threshold (descending,
//    stable ties by column) element via 4x8-bit radix select on
//    order-preserving keys computed on the fly from the LDS floats
//  - fin = min(both)
// ---------------------------------------------------------------------------
__global__ __launch_bounds__(256) void cm_select_kernel(
    const float* __restrict__ Dmat, const int* __restrict__ taup,
    int* __restrict__ fin) {
  __shared__ __align__(16) float svals[BN];   // 32 KB of the 320 KB/WGP
  __shared__ unsigned hist[256];
  __shared__ int   sscan[256];
  __shared__ float sgap[256];
  __shared__ int   sgidx[256];
  __shared__ unsigned s_prefix;
  __shared__ int s_rank, s_ecount, s_negidx;

  const int row = blockIdx.x;
  const int t   = threadIdx.x;
  const float* grow = Dmat + (size_t)row * BN;
  const float diag  = grow[row];

  // Async copy of the whole row into LDS: 2048 x b128, 8 per thread.
  for (int j = t * 4; j < BN; j += 1024) {
    __builtin_amdgcn_global_load_async_to_lds_b128(
        (gv4i_p)(grow + j), (lv4i_p)(svals + j), 0, 0);
  }
#if __has_builtin(__builtin_amdgcn_s_wait_asynccnt)
  __builtin_amdgcn_s_wait_asynccnt(0);
#else
  asm volatile("s_wait_asynccnt 0" ::: "memory");
#endif
  __syncthreads();

  if (t == 0) {
    svals[row] = -1.0f;  // rev_dists diagonal
    // idx_threshold = max(round_half_even(tau*(B-1)) - 1, 0); asc rank = B-1-it
    const double tau = (double)taup[0] / 100.0;
    int it = __double2int_rn(tau * (double)(BN - 1)) - 1;
    if (it < 0) it = 0;
    s_rank   = (BN - 1) - it;
    s_prefix = 0u;
    s_ecount = 0;
    s_negidx = 0;
  }
  __syncthreads();

  // Gap argmin (first-occurrence semantics -> lexicographic on (gap, j)).
  float bg = __builtin_inff();
  int   bj = 0x7fffffff;
  for (int j = t; j < BN; j += 256) {
    const float v = svals[j];
    const float g = (j == row) ? __builtin_inff() : fabsf(v - diag);
    if (g < bg || (g == bg && j < bj)) { bg = g; bj = j; }
  }
  sgap[t] = bg; sgidx[t] = bj;
  __syncthreads();
  for (int off = 128; off > 0; off >>= 1) {
    if (t < off) {
      const float og = sgap[t + off];
      const int   oj = sgidx[t + off];
      if (og < sgap[t] || (og == sgap[t] && oj < sgidx[t])) {
        sgap[t] = og; sgidx[t] = oj;
      }
    }
    __syncthreads();
  }
  const int hsn = sgidx[0];

  // 4-pass 8-bit radix select for the ascending-rank s_rank key.
  for (int sh = 24; sh >= 0; sh -= 8) {
    hist[t] = 0;
    __syncthreads();
    const unsigned pref = s_prefix;
    for (int j = t; j < BN; j += 256) {
      const unsigned k = fkey(svals[j]);
      const bool m = (sh == 24) || (((k ^ pref) >> (sh + 8)) == 0);
      if (m) atomicAdd(&hist[(k >> sh) & 255u], 1u);
    }
    __syncthreads();
    if (t == 0) {
      unsigned cum = 0;
      for (int b = 0; b < 256; ++b) {
        const unsigned h = hist[b];
        if ((unsigned)s_rank < cum + h) {
          s_rank   -= (int)cum;
          s_prefix |= ((unsigned)b) << sh;
          s_ecount  = (int)h;
          break;
        }
        cum += h;
      }
    }
    __syncthreads();
  }

  const unsigned vkey = s_prefix;
  // Descending stable sort ranks equal keys by ascending column; we need
  // column rank m = e - 1 - (ascending rank within the equal group).
  const int m = s_ecount - 1 - s_rank;

  int cnt = 0;
  const int jb = t * (BN / 256);
  for (int j = jb; j < jb + BN / 256; ++j) cnt += (fkey(svals[j]) == vkey);
  sscan[t] = cnt;
  __syncthreads();
  if (t == 0) {
    int run = 0;
    for (int i = 0; i < 256; ++i) { const int c = sscan[i]; sscan[i] = run; run += c; }
  }
  __syncthreads();
  const int pf = sscan[t];
  if (m >= pf && m < pf + cnt) {
    int need = m - pf;
    for (int j = jb; j < jb + BN / 256; ++j) {
      if (fkey(svals[j]) == vkey) {
        if (need == 0) { s_negidx = j; break; }
        --need;
      }
    }
  }
  __syncthreads();
  if (t == 0) fin[row] = (hsn < s_negidx) ? hsn : s_negidx;
}

// ---------------------------------------------------------------------------
// Output: out[0:BN*DK] = teacher; out[BN*DK:] = student[fin_idx]
// ---------------------------------------------------------------------------
__global__ __launch_bounds__(256) void cm_writeout_kernel(
    const float* __restrict__ T, const float* __restrict__ S,
    const int* __restrict__ fin, float* __restrict__ out) {
  const int row = blockIdx.x;
  const int c   = threadIdx.x * 4;
  const size_t ro = (size_t)row * DK + c;
  const float4 tv = *(const float4*)(T + ro);
  *(float4*)(out + ro) = tv;
  const int fr = fin[row];
  const float4 sv = *(const float4*)(S + (size_t)fr * DK + c);
  *(float4*)(out + (size_t)BN * DK + ro) = sv;
}

extern "C" void kernel_launch(void* const* d_in, const int* in_sizes, int n_in,
                              void* d_out, int out_size, void* d_ws, size_t ws_size,
                              hipStream_t stream) {
  const float* T    = (const float*)d_in[0];
  const float* S    = (const float*)d_in[1];
  const int*   taup = (const int*)d_in[2];
  float* out = (float*)d_out;

  float* Dmat = (float*)d_ws;                 // BN*BN f32 = 256 MB
  float* x2   = Dmat + (size_t)BN * BN;       // BN f32
  float* y2   = x2 + BN;                      // BN f32
  int*   fin  = (int*)(y2 + BN);              // BN i32

  cm_rownorm_kernel<<<BN / 8, 256, 0, stream>>>(T, x2);
  cm_rownorm_kernel<<<BN / 8, 256, 0, stream>>>(S, y2);

  dim3 grid(BN / 128, BN / 64);
  cm_dist_gemm_kernel<<<grid, 256, 0, stream>>>(T, S, x2, y2, Dmat);

  cm_select_kernel<<<BN, 256, 0, stream>>>(Dmat, taup, fin);
  cm_writeout_kernel<<<BN, 256, 0, stream>>>(T, S, fin, out);
}